// Attention_41669772705844
// MI455X (gfx1250) — compile-verified
//
#include <hip/hip_runtime.h>

typedef float v2f __attribute__((ext_vector_type(2)));
typedef float v8f __attribute__((ext_vector_type(8)));

#define B_    32
#define H_    32
#define HKV_  8
#define G_    4
#define D_    128
#define BS_   16
#define MAXB_ 256
#define SCALE_ 0.08838834764831845f
#define PSTRIDE_ 520   // per-partial floats: 512 acc + 4 M + 4 L

// Split-K paged attention. Grid = 256*nsplit workgroups: bkv = blockIdx.x%256,
// split = blockIdx.x/256. Each WG runs 8 waves of online-softmax over its
// contiguous share of token blocks; partials (M, L, U) go to ws (or, in
// direct mode with nsplit==1, normalized output goes straight to out).
__global__ __launch_bounds__(256)
void paged_attn_wmma(const float* __restrict__ q,
                     const float* __restrict__ k,
                     const float* __restrict__ v,
                     const float* __restrict__ k_cache,
                     const float* __restrict__ v_cache,
                     const int*   __restrict__ slot_mapping,
                     const int*   __restrict__ block_tables,
                     const int*   __restrict__ context_lens,
                     float*       __restrict__ out,
                     float*       __restrict__ ws,
                     int nsplit, int direct)
{
  const int tid  = threadIdx.x;
  const int lane = tid & 31;
  const int w    = tid >> 5;          // wave id 0..7
  const int half = (lane >> 4) & 1;   // lane half (0: lanes 0-15, 1: 16-31)
  const int m16  = lane & 15;         // token-row / head-col index inside tile
  const int bkv  = blockIdx.x & 255;
  const int split= blockIdx.x >> 8;
  const int b    = bkv >> 3;
  const int kv   = bkv & 7;
  const float NEGINF = -__builtin_inff();

  __shared__ float q_sh[16 * D_];              // Q^T source, heads 4..15 zeroed, pre-scaled
  __shared__ float fin_sh[G_ * D_];            // cross-wave output accumulation
  __shared__ float red_m[8 * 16];              // per-wave per-head running max
  __shared__ float red_l[8 * 16];              // per-wave per-head running sum
  __shared__ __align__(16) int sm_sh[32];      // slot_mapping
  __shared__ const float* vrow[8][16];         // per-wave V row pointers for current block

  float* wp = ws + (size_t)blockIdx.x * PSTRIDE_;
  const int ctx = __builtin_amdgcn_readfirstlane(context_lens[b]);

  if (ctx <= 0) {   // has_tokens == false -> zero result (uniform across WG)
    if (direct) {
      for (int i = tid; i < G_ * D_; i += 256)
        out[(size_t)(b * H_ + kv * G_) * D_ + i] = 0.0f;
    } else {
      for (int i = tid; i < G_ * D_; i += 256) wp[i] = 0.0f;
      if (tid < G_) { wp[512 + tid] = NEGINF; wp[516 + tid] = 0.0f; }
    }
    return;
  }

  for (int i = tid; i < 16 * D_; i += 256) {
    const int g = i >> 7, d = i & (D_ - 1);
    q_sh[i] = (g < G_) ? q[(size_t)(b * H_ + kv * G_ + g) * D_ + d] * SCALE_ : 0.0f;
  }
  for (int i = tid; i < G_ * D_; i += 256) fin_sh[i] = 0.0f;
  if (tid < 32) sm_sh[tid] = slot_mapping[tid];
  __syncthreads();

  const v8f zero8 = {0.f, 0.f, 0.f, 0.f, 0.f, 0.f, 0.f, 0.f};
  v8f acc[8];                       // out^T: rows = dims (8 slices of 16), cols = heads
  #pragma unroll
  for (int s = 0; s < 8; ++s) acc[s] = zero8;
  float m_run = NEGINF;
  float l_run = 0.0f;

  // contiguous, balanced block range for this split
  const int nblk = (ctx + BS_ - 1) >> 4;
  const int lo = (int)(((long)nblk * split) / nsplit);
  const int hi = (int)(((long)nblk * (split + 1)) / nsplit);

  for (int blk = lo + w; blk < hi; blk += 8) {
    const int bt = __builtin_amdgcn_readfirstlane(block_tables[(size_t)b * MAXB_ + blk]);
    if (bt < 0) continue;           // masked block (block_mask), wave-uniform

    // ---- virtual KV-cache scatter: redirect rows hit by slot_mapping ----
    const int slot = bt * BS_ + m16;
    int jm = -1;
    #pragma unroll
    for (int j4 = 0; j4 < 8; ++j4) {           // last write wins
      const int4 smq = *(const int4*)(sm_sh + 4 * j4);
      if (smq.x == slot) jm = 4 * j4 + 0;
      if (smq.y == slot) jm = 4 * j4 + 1;
      if (smq.z == slot) jm = 4 * j4 + 2;
      if (smq.w == slot) jm = 4 * j4 + 3;
    }
    const float* kb; const float* vb;
    if (jm >= 0) {
      kb = k + (size_t)(jm * HKV_ + kv) * D_;
      vb = v + (size_t)(jm * HKV_ + kv) * D_;
    } else {
      const size_t t = (size_t)bt * BS_ + m16;
      kb = k_cache + (t * HKV_ + kv) * D_;
      vb = v_cache + (t * HKV_ + kv) * D_;
    }
    if (lane < 16) vrow[w][m16] = vb;
    asm volatile("s_wait_dscnt 0x0" ::: "memory");

    // ---- batched K-tile loads: deep global_load_b64 pipeline per wave ----
    v2f kf[32];
    #pragma unroll
    for (int i = 0; i < 32; ++i)
      kf[i] = *(const v2f*)(kb + 4 * i + 2 * half);

    // ---- batched V-tile loads, issued now, consumed after softmax ----
    float va0[4][8], va1[4][8];
    #pragma unroll
    for (int j = 0; j < 4; ++j) {
      const int t = 4 * j + 2 * half;
      const float* r0 = vrow[w][t];
      const float* r1 = vrow[w][t + 1];
      #pragma unroll
      for (int s = 0; s < 8; ++s) {
        va0[j][s] = r0[s * 16 + m16];
        va1[j][s] = r1[s * 16 + m16];
      }
    }

    // ---- scores^T = K_tile x Q^T : two interleaved WMMA chains ----
    v8f sc0 = zero8, sc1 = zero8;
    #pragma unroll
    for (int i = 0; i < 32; i += 2) {
      const v2f q0 = *(const v2f*)(q_sh + m16 * D_ + 4 * i       + 2 * half);
      const v2f q1 = *(const v2f*)(q_sh + m16 * D_ + 4 * (i + 1) + 2 * half);
      sc0 = __builtin_amdgcn_wmma_f32_16x16x4_f32(false, kf[i],     false, q0, (short)0, sc0, false, false);
      sc1 = __builtin_amdgcn_wmma_f32_16x16x4_f32(false, kf[i + 1], false, q1, (short)0, sc1, false, false);
    }
    const v8f scv = sc0 + sc1;

    // ---- masking + online softmax (head = m16, tokens in scv[0..7]) ----
    float x[8];
    const int base = blk * BS_ + half * 8;
    #pragma unroll
    for (int r = 0; r < 8; ++r)
      x[r] = ((base + r) < ctx) ? scv[r] : NEGINF;
    float bm = x[0];
    #pragma unroll
    for (int r = 1; r < 8; ++r) bm = fmaxf(bm, x[r]);
    bm = fmaxf(bm, __shfl_xor(bm, 16, 32));
    const float mn    = fmaxf(m_run, bm);
    const float alpha = __expf(m_run - mn);
    float p[8]; float ls = 0.0f;
    #pragma unroll
    for (int r = 0; r < 8; ++r) { p[r] = __expf(x[r] - mn); ls += p[r]; }
    ls += __shfl_xor(ls, 16, 32);
    l_run = l_run * alpha + ls;
    m_run = mn;
    #pragma unroll
    for (int s = 0; s < 8; ++s) acc[s] = acc[s] * alpha;

    // ---- acc^T += V^T x p^T : 4 token-quads x 8 dim-slices ----
    #pragma unroll
    for (int j = 0; j < 4; ++j) {
      float b0, b1;
      if (j < 2) {                                     // tokens 4j..4j+3 in lower rows
        const float s0 = __shfl_xor(p[4*j+2], 16, 32);
        const float s1 = __shfl_xor(p[4*j+3], 16, 32);
        b0 = (lane < 16) ? p[4*j]   : s0;
        b1 = (lane < 16) ? p[4*j+1] : s1;
      } else {                                         // tokens 8..15 live in upper lanes
        const int r0 = 4*j - 8;
        const float s0 = __shfl_xor(p[r0],   16, 32);
        const float s1 = __shfl_xor(p[r0+1], 16, 32);
        b0 = (lane < 16) ? s0 : p[r0+2];
        b1 = (lane < 16) ? s1 : p[r0+3];
      }
      v2f pb; pb.x = b0; pb.y = b1;                    // B: (token rows, head cols)
      #pragma unroll
      for (int s = 0; s < 8; ++s) {
        v2f av; av.x = va0[j][s]; av.y = va1[j][s];    // A: (dim col, tokens t,t+1)
        acc[s] = __builtin_amdgcn_wmma_f32_16x16x4_f32(false, av, false, pb, (short)0, acc[s], false, false);
      }
    }
    asm volatile("" ::: "memory");
  }

  // ---- combine across the 8 waves of this WG ----
  if (lane < 16) { red_m[w*16 + lane] = m_run; red_l[w*16 + lane] = l_run; }
  __syncthreads();

  float M = NEGINF;
  #pragma unroll
  for (int w2 = 0; w2 < 8; ++w2) M = fmaxf(M, red_m[w2*16 + m16]);
  const float self_scale = (m_run == NEGINF) ? 0.0f : __expf(m_run - M);  // empty-wave guard
  if (m16 < G_) {
    #pragma unroll
    for (int s = 0; s < 8; ++s)
      #pragma unroll
      for (int r = 0; r < 8; ++r) {
        const int d = s*16 + r + half*8;
        atomicAdd(&fin_sh[m16 * D_ + d], acc[s][r] * self_scale);
      }
  }
  __syncthreads();

  if (direct) {
    #pragma unroll
    for (int it = 0; it < 2; ++it) {
      const int idx = tid + it * 256;       // 512 outputs = 4 heads x 128 dims
      const int h = idx >> 7;
      const int d = idx & (D_ - 1);
      float M2 = NEGINF;
      #pragma unroll
      for (int w2 = 0; w2 < 8; ++w2) M2 = fmaxf(M2, red_m[w2*16 + h]);
      float val = 0.0f;
      if (M2 != NEGINF) {
        float L = 0.0f;
        #pragma unroll
        for (int w2 = 0; w2 < 8; ++w2) L += __expf(red_m[w2*16 + h] - M2) * red_l[w2*16 + h];
        val = fin_sh[h * D_ + d] / L;
      }
      out[(size_t)(b * H_ + kv * G_ + h) * D_ + d] = val;
    }
  } else {
    #pragma unroll
    for (int it = 0; it < 2; ++it) {
      const int idx = tid + it * 256;
      wp[idx] = fin_sh[idx];                // unnormalized exp-weighted sums
    }
    if (tid < G_) {                         // per-head M and L for this partial
      const int h = tid;
      float M2 = NEGINF;
      #pragma unroll
      for (int w2 = 0; w2 < 8; ++w2) M2 = fmaxf(M2, red_m[w2*16 + h]);
      float L = 0.0f;
      if (M2 != NEGINF) {
        #pragma unroll
        for (int w2 = 0; w2 < 8; ++w2) L += __expf(red_m[w2*16 + h] - M2) * red_l[w2*16 + h];
      }
      wp[512 + h] = M2;
      wp[516 + h] = L;
    }
  }
}

// Log-sum-exp merge of the nsplit partials; 256 WGs, one per (b, kv).
__global__ __launch_bounds__(256)
void paged_attn_combine(const float* __restrict__ ws,
                        float* __restrict__ out,
                        int nsplit)
{
  const int tid = threadIdx.x;
  const int bkv = blockIdx.x;
  const int b   = bkv >> 3;
  const int kv  = bkv & 7;
  const float NEGINF = -__builtin_inff();

  #pragma unroll
  for (int it = 0; it < 2; ++it) {
    const int idx = tid + it * 256;        // 512 outputs = 4 heads x 128 dims
    const int h = idx >> 7;
    const int d = idx & (D_ - 1);
    float M = NEGINF;
    for (int s = 0; s < nsplit; ++s)
      M = fmaxf(M, ws[(size_t)(s * 256 + bkv) * PSTRIDE_ + 512 + h]);
    float val = 0.0f;
    if (M != NEGINF) {                     // else: no valid tokens anywhere -> 0
      float L = 0.0f, U = 0.0f;
      for (int s = 0; s < nsplit; ++s) {
        const float* p = ws + (size_t)(s * 256 + bkv) * PSTRIDE_;
        const float sc = __expf(p[512 + h] - M);   // exp(-inf)=0 for empty splits
        L += sc * p[516 + h];
        U += sc * p[idx];
      }
      val = U / L;
    }
    out[(size_t)(b * H_ + kv * G_ + h) * D_ + d] = val;
  }
}

extern "C" void kernel_launch(void* const* d_in, const int* in_sizes, int n_in,
                              void* d_out, int out_size, void* d_ws, size_t ws_size,
                              hipStream_t stream) {
  (void)in_sizes; (void)n_in; (void)out_size;
  const float* q  = (const float*)d_in[0];
  const float* k  = (const float*)d_in[1];
  const float* v  = (const float*)d_in[2];
  const float* kc = (const float*)d_in[3];
  const float* vc = (const float*)d_in[4];
  const int*   sm = (const int*)d_in[5];
  const int*   bt = (const int*)d_in[6];
  const int*   cl = (const int*)d_in[7];
  float* out = (float*)d_out;
  float* ws  = (float*)d_ws;

  // Deterministic split-count selection from the available workspace.
  const size_t per_split_bytes = (size_t)256 * PSTRIDE_ * sizeof(float);
  int nsplit = 4;
  if (ws_size < 4 * per_split_bytes) nsplit = 2;
  if (ws_size < 2 * per_split_bytes) nsplit = 1;
  const int direct = (ws_size < per_split_bytes) ? 1 : 0;
  if (direct) nsplit = 1;

  dim3 block(256);
  dim3 grid1(256 * nsplit);
  paged_attn_wmma<<<grid1, block, 0, stream>>>(q, k, v, kc, vc, sm, bt, cl,
                                               out, ws, nsplit, direct);
  if (!direct) {
    dim3 grid2(256);
    paged_attn_combine<<<grid2, block, 0, stream>>>(ws, out, nsplit);
  }
}